// BatchMultiHeadGraphAttention_4217657885343
// MI455X (gfx1250) — compile-verified
//
#include <hip/hip_runtime.h>
#include <math.h>

typedef __attribute__((ext_vector_type(2))) float v2f;
typedef __attribute__((ext_vector_type(8))) float v8f;

#define NB 8
#define NH 8
#define NN 1024
#define NF 64
#define NEG_SLOPE 0.2f

// ---------------------------------------------------------------------------
// Kernel A: h_prime[b,h,n,o] = sum_f h[b,n,f] * w[h,f,o]
// One wave32 per 16x16 output tile, K=64 via 16 x V_WMMA_F32_16X16X4_F32.
// A (16x4 f32): lanes 0-15 hold M=0..15 K=(k0,k0+1); lanes 16-31 K=(k0+2,k0+3).
// B (4x16 f32): mirrored; N = lane&15.
// C/D (16x16 f32): 8 VGPRs, M = r + 8*(lane>=16), N = lane&15.
// ---------------------------------------------------------------------------
__global__ __launch_bounds__(256) void gat_hprime_wmma(
    const float* __restrict__ h, const float* __restrict__ w,
    float* __restrict__ hp)
{
    const int lane = threadIdx.x & 31;
    const int wave = threadIdx.x >> 5;
    int tile = blockIdx.x * 8 + wave;
    const int co = tile & 3;  tile >>= 2;   // 4 col tiles  (NF/16)
    const int ro = tile & 63; tile >>= 6;   // 64 row tiles (NN/16)
    const int hd = tile & 7;  tile >>= 3;
    const int b  = tile;

    const int rbase = ro * 16, cbase = co * 16;
    const int m     = lane & 15;
    const int khalf = (lane >> 4) << 1;     // 0 or 2

    const float* hA = h + ((size_t)b * NN + rbase) * NF;   // 16 rows of h
    const float* wB = w + (size_t)hd * NF * NF;

    v8f c = {0.f, 0.f, 0.f, 0.f, 0.f, 0.f, 0.f, 0.f};
    #pragma unroll 4
    for (int k0 = 0; k0 < NF; k0 += 4) {
        const int kb = k0 + khalf;
        v2f a, bm;
        a.x  = hA[m * NF + kb];
        a.y  = hA[m * NF + kb + 1];
        bm.x = wB[kb * NF + cbase + m];
        bm.y = wB[(kb + 1) * NF + cbase + m];
        c = __builtin_amdgcn_wmma_f32_16x16x4_f32(
                /*neg_a=*/false, a, /*neg_b=*/false, bm,
                /*c_mod=*/(short)0, c, /*reuse_a=*/false, /*reuse_b=*/false);
    }

    float* outp = hp + (((size_t)b * NH + hd) * NN + rbase) * NF + cbase;
    const int rshift = (lane >> 4) << 3;    // 0 or 8
    #pragma unroll
    for (int r = 0; r < 8; ++r)
        outp[(r + rshift) * NF + m] = c[r];
}

// ---------------------------------------------------------------------------
// Kernel A2: attn_src[b,h,n] = h_prime[b,h,n,:] . a_src[h,:],  same for dst.
// One thread per (b,h,n) row; 64-wide dot, coalesced across consecutive rows.
// ---------------------------------------------------------------------------
__global__ __launch_bounds__(256) void gat_scores(
    const float* __restrict__ hp, const float* __restrict__ a_src,
    const float* __restrict__ a_dst,
    float* __restrict__ s_src, float* __restrict__ s_dst)
{
    const int idx = blockIdx.x * blockDim.x + threadIdx.x;   // (b*NH+h)*NN + n
    const int hd  = (idx / NN) & (NH - 1);
    const float* row = hp + (size_t)idx * NF;
    const float* as  = a_src + hd * NF;
    const float* ad  = a_dst + hd * NF;
    float s = 0.f, d = 0.f;
    #pragma unroll 8
    for (int o = 0; o < NF; ++o) {
        const float v = row[o];
        s += v * as[o];
        d += v * ad[o];
    }
    s_src[idx] = s;
    s_dst[idx] = d;
}

// ---------------------------------------------------------------------------
// Kernel B: attn row softmax. One wave32 per row (b,h,i): 32 elems per lane
// held in registers; leaky_relu + Relation mask fused; wave32 shuffle
// reductions for max and sum; coalesced 4KB row write (HBM-floor kernel).
// ---------------------------------------------------------------------------
__global__ __launch_bounds__(256) void gat_softmax(
    const float* __restrict__ Rel, const float* __restrict__ s_src,
    const float* __restrict__ s_dst, float* __restrict__ attn)
{
    const int lane = threadIdx.x & 31;
    const int wave = threadIdx.x >> 5;
    const int row  = blockIdx.x * 8 + wave;     // (b*NH+h)*NN + i
    const int i    = row & (NN - 1);
    const int bh   = row >> 10;
    const int b    = bh >> 3;

    const float  si   = s_src[row];
    const float* dvec = s_dst + (size_t)bh * NN;
    const float* rrow = Rel + ((size_t)b * NN + i) * NN;

    float e[32];
    float mx = -INFINITY;
    #pragma unroll
    for (int t = 0; t < 32; ++t) {
        const int j = lane + 32 * t;
        float x = si + dvec[j];
        x = (x >= 0.f) ? x : NEG_SLOPE * x;     // leaky_relu(0.2)
        x *= rrow[j];                           // relation mask
        e[t] = x;
        mx = fmaxf(mx, x);
    }
    #pragma unroll
    for (int off = 16; off > 0; off >>= 1)
        mx = fmaxf(mx, __shfl_xor(mx, off, 32));

    float s = 0.f;
    #pragma unroll
    for (int t = 0; t < 32; ++t) { e[t] = __expf(e[t] - mx); s += e[t]; }
    #pragma unroll
    for (int off = 16; off > 0; off >>= 1)
        s += __shfl_xor(s, off, 32);

    const float inv = 1.f / s;
    float* orow = attn + (size_t)row * NN;
    #pragma unroll
    for (int t = 0; t < 32; ++t)
        orow[lane + 32 * t] = e[t] * inv;
}

// ---------------------------------------------------------------------------
// Kernel C: output[b,h,i,o] = sum_j attn[b,h,i,j] * h_prime[b,h,j,o] + bias[o]
// One wave32 per 16x16 output tile, K=1024 via 256 x V_WMMA_F32_16X16X4_F32.
// ---------------------------------------------------------------------------
__global__ __launch_bounds__(256) void gat_output_wmma(
    const float* __restrict__ attn, const float* __restrict__ hp,
    const float* __restrict__ bias, float* __restrict__ out)
{
    const int lane = threadIdx.x & 31;
    const int wave = threadIdx.x >> 5;
    int tile = blockIdx.x * 8 + wave;
    const int co = tile & 3;  tile >>= 2;
    const int ro = tile & 63; tile >>= 6;
    const int bh = tile;                        // b*NH + h

    const int rbase = ro * 16, cbase = co * 16;
    const int m     = lane & 15;
    const int khalf = (lane >> 4) << 1;

    const float* Arow = attn + ((size_t)bh * NN + rbase) * NN;  // 16 x 1024
    const float* Bmat = hp + (size_t)bh * NN * NF;              // 1024 x 64

    v8f c = {0.f, 0.f, 0.f, 0.f, 0.f, 0.f, 0.f, 0.f};
    for (int k0 = 0; k0 < NN; k0 += 4) {
        const int kb = k0 + khalf;
        v2f a, bm;
        a.x  = Arow[m * NN + kb];
        a.y  = Arow[m * NN + kb + 1];
        bm.x = Bmat[kb * NF + cbase + m];
        bm.y = Bmat[(kb + 1) * NF + cbase + m];
        c = __builtin_amdgcn_wmma_f32_16x16x4_f32(
                false, a, false, bm, (short)0, c, false, false);
    }

    const float bcol = bias[cbase + m];
    float* op = out + ((size_t)bh * NN + rbase) * NF + cbase;
    const int rshift = (lane >> 4) << 3;
    #pragma unroll
    for (int r = 0; r < 8; ++r)
        op[(r + rshift) * NF + m] = c[r] + bcol;
}

// ---------------------------------------------------------------------------
extern "C" void kernel_launch(void* const* d_in, const int* in_sizes, int n_in,
                              void* d_out, int out_size, void* d_ws, size_t ws_size,
                              hipStream_t stream) {
    const float* h     = (const float*)d_in[0];   // [8,1024,64]
    const float* Rel   = (const float*)d_in[1];   // [8,1024,1024]
    const float* w     = (const float*)d_in[2];   // [8,64,64]
    const float* a_src = (const float*)d_in[3];   // [8,64,1]
    const float* a_dst = (const float*)d_in[4];   // [8,64,1]
    const float* bias  = (const float*)d_in[5];   // [64]

    float* out  = (float*)d_out;                          // [8,8,1024,64]
    float* attn = out + (size_t)NB * NH * NN * NF;        // [8,8,1024,1024]

    // workspace: h_prime (16.8 MB) + src/dst score vectors (0.5 MB)
    float* hp    = (float*)d_ws;
    float* s_src = hp + (size_t)NB * NH * NN * NF;
    float* s_dst = s_src + (size_t)NB * NH * NN;

    const int tiles = NB * NH * (NN / 16) * (NF / 16);    // 16384 wave-tiles

    gat_hprime_wmma<<<tiles / 8, 256, 0, stream>>>(h, w, hp);
    gat_scores<<<(NB * NH * NN) / 256, 256, 0, stream>>>(hp, a_src, a_dst, s_src, s_dst);
    gat_softmax<<<(NB * NH * NN) / 8, 256, 0, stream>>>(Rel, s_src, s_dst, attn);
    gat_output_wmma<<<tiles / 8, 256, 0, stream>>>(attn, hp, bias, out);
}